// KNNComputer_40604620817221
// MI455X (gfx1250) — compile-verified
//
#include <hip/hip_runtime.h>
#include <hip/hip_bf16.h>
#include <math.h>
#include <stdint.h>

typedef __attribute__((ext_vector_type(2))) float v2f;
typedef __attribute__((ext_vector_type(8))) float v8f;

#define DIMS 1024      // feature dim
#define BX   2048      // x rows
#define MY   8192      // y rows
#define KNN  8
#define TM   16        // x rows per block
#define TN   128       // y cols per strip (4 waves x 2 tiles x 16)
#define KC   128       // k chunk staged in LDS
#define NTHREADS 128
#define FBIG 3.0e38f

// LDS layout inside ONE allocation (placement guaranteed):
//   [0x00000) y chunk   TN*KC   floats (64 KB)  <- B offsets fit DS imm field
//   [0x10000) x tile    TM*DIMS floats (64 KB)
//   [0x20000) dist tile TN*TM   floats ( 8 KB)
//   [0x22000) x norms   TM      floats
#define OFF_Y 0
#define OFF_X (TN * KC)
#define OFF_D (OFF_X + TM * DIMS)
#define OFF_N (OFF_D + TN * TM)
#define SMEM_FLOATS (OFF_N + TM)

// ---- CDNA5 async global->LDS copy (ASYNCcnt-tracked, no VGPR staging) -----
__device__ __forceinline__ void async_ld_b128(uint32_t lds_addr, const float* gptr) {
  asm volatile("global_load_async_to_lds_b128 %0, %1, off"
               :: "v"(lds_addr), "v"(gptr)
               : "memory");
}
__device__ __forceinline__ void wait_asynccnt0() {
  asm volatile("s_wait_asynccnt 0x0" ::: "memory");
}

// ---------------- Kernel A: pass through the persistent min_dists buffer ---
__global__ void copy_min_dists(const float* __restrict__ src,
                               float* __restrict__ dst, int n) {
  int i = blockIdx.x * blockDim.x + threadIdx.x;
  if (i < n) dst[i] = src[i];
}

// ---------------- Kernel B: squared row norms for x and y -------------------
__global__ void row_norms(const float* __restrict__ x, const float* __restrict__ y,
                          float* __restrict__ x2, float* __restrict__ y2) {
  const int wave = threadIdx.x >> 5;
  const int lane = threadIdx.x & 31;
  const int row  = blockIdx.x * (blockDim.x >> 5) + wave;
  const float* src;
  float* dst;
  int r;
  if (row < BX) { src = x; dst = x2; r = row; }
  else          { src = y; dst = y2; r = row - BX; }
  const float* p = src + (size_t)r * DIMS;
  float s = 0.0f;
  #pragma unroll
  for (int i = 0; i < DIMS; i += 128) {
    float4 v = *(const float4*)(p + i + lane * 4);
    s += v.x * v.x + v.y * v.y + v.z * v.z + v.w * v.w;
  }
  #pragma unroll
  for (int m = 16; m > 0; m >>= 1) s += __shfl_xor(s, m, 32);
  if (lane == 0) dst[r] = s;
}

// ---------------- Kernel C: WMMA GEMM tiles + running top-8 merge -----------
// K-slot/data-column assignment (A and B must agree):
//   slot0<-c0, slot1<-c2, slot2<-c1, slot3<-c3
// B frag (lanes0-15 slots {0,2}, lanes16-31 slots {1,3}) reads UNPERMUTED
// lds_y at [k0+2*hi, k0+2*hi+1] -> single b64, async-DMA friendly.
// A frag (lanes0-15 slots {0,1}, lanes16-31 slots {2,3}) reads lds_x stored
// K-permuted per 4-group as {c0,c2,c1,c3} -> single b64.
__global__ void __launch_bounds__(NTHREADS)
knn_topk_kernel(const float* __restrict__ x, const float* __restrict__ y,
                const float* __restrict__ mdin,
                const int* __restrict__ xsp, const int* __restrict__ ysp,
                const float* __restrict__ x2g, const float* __restrict__ y2g,
                float* __restrict__ out) {
  __shared__ __align__(16) float smem[SMEM_FLOATS];
  float* lds_y = smem + OFF_Y;
  float* lds_x = smem + OFF_X;
  float* lds_d = smem + OFF_D;
  float* lx2   = smem + OFF_N;

  const int tid  = threadIdx.x;
  const int lane = tid & 31;
  const int wave = tid >> 5;
  const int hi   = lane >> 4;    // half-wave selector (ISA frag layouts)
  const int l16  = lane & 15;
  const int xs   = xsp[0];
  const int ys   = ysp[0];
  const int x0   = blockIdx.x * TM;

  // ---- stage x tile (16 x 1024 f32) to LDS with per-4 K permute ----
  {
    const float4* gx = (const float4*)(x + (size_t)x0 * DIMS);
    float4* sx = (float4*)lds_x;
    #pragma unroll 4
    for (int i = tid; i < TM * DIMS / 4; i += NTHREADS) {
      float4 v = gx[i];
      float4 p = make_float4(v.x, v.z, v.y, v.w);   // {c0,c2,c1,c3}
      sx[i] = p;
    }
  }
  if (tid < TM) lx2[tid] = x2g[x0 + tid];

  // ---- seed per-row top-8 from the running buffer (ascending) ----
  float best[KNN];
  if (tid < TM) {
    #pragma unroll
    for (int j = 0; j < KNN; ++j) best[j] = mdin[(size_t)(xs + x0 + tid) * KNN + j];
    #pragma unroll
    for (int a = 1; a < KNN; ++a) {
      float v = best[a];
      int b = a - 1;
      while (b >= 0 && best[b] > v) { best[b + 1] = best[b]; --b; }
      best[b + 1] = v;
    }
  } else {
    #pragma unroll
    for (int j = 0; j < KNN; ++j) best[j] = FBIG;
  }
  __syncthreads();

  const int col0 = wave * 16 + l16;       // strip-local col, tile 0
  const int col1 = col0 + 64;             // strip-local col, tile 1

  for (int s = 0; s < MY / TN; ++s) {
    const int n0 = s * TN;                // strip base y row
    v8f acc0 = {};
    v8f acc1 = {};

    for (int kc = 0; kc < DIMS / KC; ++kc) {
      __syncthreads();                    // previous chunk fully consumed
      // async DMA y chunk [TN rows][KC floats] -> LDS (4096 b128, 32/thread)
      for (int i = tid; i < TN * KC / 4; i += NTHREADS) {
        const int row = i >> 5;           // KC/4 = 32 float4 per row
        const int c4  = i & 31;
        async_ld_b128((uint32_t)(uintptr_t)&lds_y[row * KC + c4 * 4],
                      y + (size_t)(n0 + row) * DIMS + kc * KC + c4 * 4);
      }
      wait_asynccnt0();
      __syncthreads();

      const float* xa  = &lds_x[(size_t)l16 * DIMS + kc * KC];
      const float* yb0 = &lds_y[(size_t)col0 * KC];
      const float* yb1 = &lds_y[(size_t)col1 * KC];
      #pragma unroll 8
      for (int k0 = 0; k0 < KC; k0 += 4) {
        v2f a  = *(const v2f*)(xa  + k0 + 2 * hi);
        v2f b0 = *(const v2f*)(yb0 + k0 + 2 * hi);
        v2f b1 = *(const v2f*)(yb1 + k0 + 2 * hi);
        acc0 = __builtin_amdgcn_wmma_f32_16x16x4_f32(
            false, a, false, b0, (short)0, acc0, false, false);
        acc1 = __builtin_amdgcn_wmma_f32_16x16x4_f32(
            false, a, false, b1, (short)0, acc1, false, false);
      }
    }

    // ---- Gram -> distance, self-pair mask, scatter to LDS tile ----
    const float y2c0 = y2g[n0 + col0];
    const float y2c1 = y2g[n0 + col1];
    const int   yg0  = ys + n0 + col0;
    const int   yg1  = ys + n0 + col1;
    #pragma unroll
    for (int r = 0; r < 8; ++r) {
      const int row = r + 8 * hi;         // C layout: vgpr r -> rows r / r+8
      const int xg  = xs + x0 + row;
      float sq0 = lx2[row] + y2c0 - 2.0f * acc0[r];
      float sq1 = lx2[row] + y2c1 - 2.0f * acc1[r];
      float d0 = sqrtf(fmaxf(sq0, 0.0f));
      float d1 = sqrtf(fmaxf(sq1, 0.0f));
      if (xg == yg0) d0 = FBIG;
      if (xg == yg1) d1 = FBIG;
      lds_d[col0 * TM + row] = d0;
      lds_d[col1 * TM + row] = d1;
    }
    __syncthreads();

    // ---- row owners fold 128 candidates into their sorted top-8 ----
    if (tid < TM) {
      for (int c = 0; c < TN; ++c) {
        float dv = lds_d[c * TM + tid];
        if (dv < best[KNN - 1]) {
          int b = KNN - 2;
          while (b >= 0 && best[b] > dv) { best[b + 1] = best[b]; --b; }
          best[b + 1] = dv;
        }
      }
    }
    __syncthreads();
  }

  if (tid < TM) {
    float* o = out + (size_t)(xs + x0 + tid) * KNN;
    #pragma unroll
    for (int j = 0; j < KNN; ++j) o[j] = best[j];
  }
}

// ---------------------------------------------------------------------------
extern "C" void kernel_launch(void* const* d_in, const int* in_sizes, int n_in,
                              void* d_out, int out_size, void* d_ws, size_t ws_size,
                              hipStream_t stream) {
  const float* x   = (const float*)d_in[0];
  const float* y   = (const float*)d_in[1];
  const float* md  = (const float*)d_in[2];
  const int*   xsp = (const int*)d_in[3];
  const int*   ysp = (const int*)d_in[4];
  float* out = (float*)d_out;

  float* x2 = (float*)d_ws;        // 2048 floats
  float* y2 = x2 + BX;             // 8192 floats

  copy_min_dists<<<(out_size + 255) / 256, 256, 0, stream>>>(md, out, out_size);
  row_norms<<<(BX + MY) / 8, 256, 0, stream>>>(x, y, x2, y2);
  knn_topk_kernel<<<BX / TM, NTHREADS, 0, stream>>>(x, y, md, xsp, ysp, x2, y2, out);
}